// WarpImage_36850819400522
// MI455X (gfx1250) — compile-verified
//
#include <hip/hip_runtime.h>

// Problem dims (fixed by the reference)
#define CC 32
#define HH 1024
#define WW 2048
#define HW (HH * WW)
#define PIX (HH * WW)

typedef float v4f __attribute__((ext_vector_type(4)));

__global__ __launch_bounds__(256) void
warp_bilinear_kernel(const float* __restrict__ image,
                     const float* __restrict__ flow,
                     float* __restrict__ out)
{
    // Each thread handles 4 consecutive pixels (same row since WW % 4 == 0).
    const int group = blockIdx.x * blockDim.x + threadIdx.x;
    const int pix0  = group * 4;
    if (pix0 >= PIX) return;

    const int h  = pix0 / WW;
    const int w0 = pix0 - h * WW;

    // Coalesced 16B flow loads (flow[0] = u-displacement, flow[1] = v-displacement)
    const v4f fu = *(const v4f*)(flow + pix0);
    const v4f fv = *(const v4f*)(flow + HW + pix0);

    float wgt[4][4];   // bilinear weights, zeroed when the corner is out of bounds
    int   off[4][4];   // clamped gather offsets within one channel plane

    #pragma unroll
    for (int j = 0; j < 4; ++j) {
        const float u = (float)(w0 + j) + fu[j];
        const float v = (float)h + fv[j];
        const float u0f = floorf(u);
        const float v0f = floorf(v);
        const float wu = u - u0f;
        const float wv = v - v0f;
        const int u0 = (int)u0f, v0 = (int)v0f;
        const int u1 = u0 + 1,   v1 = v0 + 1;

        const bool vu0 = (u0 >= 0) & (u0 <= WW - 1);
        const bool vu1 = (u1 >= 0) & (u1 <= WW - 1);
        const bool vv0 = (v0 >= 0) & (v0 <= HH - 1);
        const bool vv1 = (v1 >= 0) & (v1 <= HH - 1);

        const int cu0 = min(max(u0, 0), WW - 1);
        const int cu1 = min(max(u1, 0), WW - 1);
        const int cv0 = min(max(v0, 0), HH - 1);
        const int cv1 = min(max(v1, 0), HH - 1);

        const float omu = 1.0f - wu;
        const float omv = 1.0f - wv;
        wgt[j][0] = (vu0 & vv0) ? omu * omv : 0.0f;
        wgt[j][1] = (vu1 & vv0) ? wu  * omv : 0.0f;
        wgt[j][2] = (vu0 & vv1) ? omu * wv  : 0.0f;
        wgt[j][3] = (vu1 & vv1) ? wu  * wv  : 0.0f;

        off[j][0] = cv0 * WW + cu0;
        off[j][1] = cv0 * WW + cu1;
        off[j][2] = cv1 * WW + cu0;
        off[j][3] = cv1 * WW + cu1;
    }

    const float* img = image;
    float*       o   = out + pix0;

    #pragma unroll 2
    for (int c = 0; c < CC; ++c) {
        // Prefetch the two corner rows for a future channel (gfx1250
        // global_prefetch_b8; speculative — dropped silently if it falls
        // past the image). Same lines the real gathers will want.
        {
            const int pfc = (c + 4 < CC) ? 4 : 0;
            const float* pf = img + pfc * HW;
            __builtin_prefetch(pf + off[0][0], 0, 1);
            __builtin_prefetch(pf + off[0][2], 0, 1);
        }

        float acc[4];
        #pragma unroll
        for (int j = 0; j < 4; ++j) {
            // 4 independent gathers per pixel; 16 loads in flight per iteration
            const float g00 = img[off[j][0]];
            const float g01 = img[off[j][1]];
            const float g10 = img[off[j][2]];
            const float g11 = img[off[j][3]];
            acc[j] = wgt[j][0] * g00 + wgt[j][1] * g01
                   + wgt[j][2] * g10 + wgt[j][3] * g11;
        }

        // Non-temporal 16B store: keep L2 free for the image read stream.
        v4f r;
        r[0] = acc[0]; r[1] = acc[1]; r[2] = acc[2]; r[3] = acc[3];
        __builtin_nontemporal_store(r, (v4f*)o);

        img += HW;
        o   += HW;
    }
}

extern "C" void kernel_launch(void* const* d_in, const int* in_sizes, int n_in,
                              void* d_out, int out_size, void* d_ws, size_t ws_size,
                              hipStream_t stream) {
    (void)in_sizes; (void)n_in; (void)out_size; (void)d_ws; (void)ws_size;
    const float* image = (const float*)d_in[0];  // (32, 1024, 2048) fp32
    const float* flow  = (const float*)d_in[1];  // (2, 1024, 2048) fp32
    float*       out   = (float*)d_out;          // (32, 1024, 2048) fp32

    const int threads = 256;
    const int groups  = PIX / 4;                 // 4 pixels per thread
    const int blocks  = (groups + threads - 1) / threads;  // 2048 blocks
    warp_bilinear_kernel<<<blocks, threads, 0, stream>>>(image, flow, out);
}